// SupCCL_27453430956190
// MI455X (gfx1250) — compile-verified
//
#include <hip/hip_runtime.h>

#define KD_T   4.0f
#define BATCH  256
#define DIM    2048
#define PNUM   56
#define TPB    256
#define VEC    8          // DIM / TPB floats per thread
#define NWAVE  (TPB / 32) // 8 waves per block (wave32)
#define PCHUNK 14         // p-rows per dcl block
#define SPLIT  4          // PNUM / PCHUNK

typedef __attribute__((ext_vector_type(2))) float v2f;
typedef __attribute__((ext_vector_type(8))) float v8f;

// ---------------- wave32 shuffle reductions --------------------------------
__device__ __forceinline__ float wave_sum(float v) {
    #pragma unroll
    for (int o = 16; o > 0; o >>= 1) v += __shfl_xor(v, o, 32);
    return v;
}
__device__ __forceinline__ float wave_max(float v) {
    #pragma unroll
    for (int o = 16; o > 0; o >>= 1) v = fmaxf(v, __shfl_xor(v, o, 32));
    return v;
}

// ---------------- symmetric-KL core over one row pair ----------------------
// kl(s,t)+kl(t,s) = T/D * (S1/Zb + S2/Zg); the log-sum-exp terms cancel.
// Returns S1/Zb + S2/Zg (valid in every thread). Deterministic (fixed order).
__device__ __forceinline__ float sym_kl_row(const float* g, const float* m,
                                            float (*shm)[2], float (*shs)[4]) {
    const float invT = 1.0f / KD_T;
    const int wid  = threadIdx.x >> 5;
    const int lane = threadIdx.x & 31;

    float lmg = -1e30f, lmb = -1e30f;
    #pragma unroll
    for (int i = 0; i < VEC; ++i) {
        lmg = fmaxf(lmg, g[i] * invT);
        lmb = fmaxf(lmb, m[i] * invT);
    }
    lmg = wave_max(lmg);
    lmb = wave_max(lmb);
    if (lane == 0) { shm[wid][0] = lmg; shm[wid][1] = lmb; }
    __syncthreads();
    float Mg = shm[0][0], Mb = shm[0][1];
    #pragma unroll
    for (int i = 1; i < NWAVE; ++i) {
        Mg = fmaxf(Mg, shm[i][0]);
        Mb = fmaxf(Mb, shm[i][1]);
    }

    float zg = 0.f, zb = 0.f, s1 = 0.f, s2 = 0.f;
    #pragma unroll
    for (int i = 0; i < VEC; ++i) {
        const float eg = __expf(g[i] * invT - Mg);
        const float eb = __expf(m[i] * invT - Mb);
        const float d  = m[i] - g[i];
        zg += eg; zb += eb;
        s1 += eb * d;
        s2 -= eg * d;
    }
    zg = wave_sum(zg); zb = wave_sum(zb);
    s1 = wave_sum(s1); s2 = wave_sum(s2);
    __syncthreads();                       // shm reads done before shs round
    if (lane == 0) { shs[wid][0] = zg; shs[wid][1] = zb; shs[wid][2] = s1; shs[wid][3] = s2; }
    __syncthreads();
    zg = zb = s1 = s2 = 0.f;
    #pragma unroll
    for (int i = 0; i < NWAVE; ++i) {
        zg += shs[i][0]; zb += shs[i][1]; s1 += shs[i][2]; s2 += shs[i][3];
    }
    return s1 / zb + s2 / zg;
}

// ---------------- async staging: one 32B/thread row slice ------------------
// INST_OFFSET is added to BOTH the LDS and global addresses (ISA 8.2), so one
// LDS-offset VGPR + one 64-bit global-addr pair covers both 16B beats.
__device__ __forceinline__ void async_issue_row(const float* src, const float* lds_dst) {
    const unsigned l0 = (unsigned)(uintptr_t)lds_dst;   // low 32 bits = LDS offset
    asm volatile(
        "s_wait_dscnt 0x0\n\t"                          // prior ds reads landed
        "global_load_async_to_lds_b128 %0, %1, off\n\t"
        "global_load_async_to_lds_b128 %0, %1, off offset:16"
        :: "v"(l0), "v"(src) : "memory");
}
__device__ __forceinline__ void async_wait_keep2() {
    asm volatile("s_wait_asynccnt 0x2" ::: "memory");   // next row still in flight
}
__device__ __forceinline__ void async_wait_all() {
    asm volatile("s_wait_asynccnt 0x0" ::: "memory");
}

// ---------------- kernel 1: ebp_bar = mean over P --------------------------
__global__ void bar_kernel(const float* __restrict__ ebp, float* __restrict__ bar) {
    const int idx = blockIdx.x * blockDim.x + threadIdx.x;   // float4 index
    const float4* src = (const float4*)ebp;
    const int n4 = BATCH * DIM / 4;
    float4 acc = make_float4(0.f, 0.f, 0.f, 0.f);
    for (int p = 0; p < PNUM; ++p) {
        float4 v = src[(size_t)p * n4 + idx];
        acc.x += v.x; acc.y += v.y; acc.z += v.z; acc.w += v.w;
    }
    const float inv = 1.0f / (float)PNUM;
    acc.x *= inv; acc.y *= inv; acc.z *= inv; acc.w *= inv;
    ((float4*)bar)[idx] = acc;
}

// ---------------- kernel 2: soft_dil partials ------------------------------
__global__ void dil_kernel(const float* __restrict__ ebg, const float* __restrict__ bar,
                           float* __restrict__ ws_dil) {
    __shared__ float shm[NWAVE][2];
    __shared__ float shs[NWAVE][4];
    const int b   = blockIdx.x;
    const int tid = threadIdx.x;
    const float4* g4 = (const float4*)(ebg + (size_t)b * DIM);
    const float4* m4 = (const float4*)(bar + (size_t)b * DIM);
    float g[VEC], m[VEC];
    float4 t;
    t = g4[tid * 2 + 0]; g[0] = t.x; g[1] = t.y; g[2] = t.z; g[3] = t.w;
    t = g4[tid * 2 + 1]; g[4] = t.x; g[5] = t.y; g[6] = t.z; g[7] = t.w;
    t = m4[tid * 2 + 0]; m[0] = t.x; m[1] = t.y; m[2] = t.z; m[3] = t.w;
    t = m4[tid * 2 + 1]; m[4] = t.x; m[5] = t.y; m[6] = t.z; m[7] = t.w;

    const float r = sym_kl_row(g, m, shm, shs);
    if (tid == 0) ws_dil[b] = r * (KD_T / (float)DIM);
}

// ---------------- kernel 3: soft_dcl (block = one b, 14 p-rows) ------------
// ebg/bar rows live in registers for all 14 rows; ebp rows are streamed
// through an LDS double buffer with GLOBAL_LOAD_ASYNC_TO_LDS_B128.
__global__ void dcl_kernel(const float* __restrict__ ebg, const float* __restrict__ bar,
                           const float* __restrict__ ebp, float* __restrict__ ws_dcl) {
    __shared__ __align__(16) float buf[2][DIM];   // 16 KB double buffer
    __shared__ float shm[NWAVE][2];
    __shared__ float shs[NWAVE][4];

    const int b   = blockIdx.x / SPLIT;
    const int p0  = (blockIdx.x % SPLIT) * PCHUNK;
    const int tid = threadIdx.x;

    const float4* g4 = (const float4*)(ebg + (size_t)b * DIM);
    const float4* m4 = (const float4*)(bar + (size_t)b * DIM);
    float g[VEC], m[VEC];
    float4 t;
    t = g4[tid * 2 + 0]; g[0] = t.x; g[1] = t.y; g[2] = t.z; g[3] = t.w;
    t = g4[tid * 2 + 1]; g[4] = t.x; g[5] = t.y; g[6] = t.z; g[7] = t.w;
    t = m4[tid * 2 + 0]; m[0] = t.x; m[1] = t.y; m[2] = t.z; m[3] = t.w;
    t = m4[tid * 2 + 1]; m[4] = t.x; m[5] = t.y; m[6] = t.z; m[7] = t.w;

    // each thread stages and consumes its own 32B slice of the row
    const float* row0 = ebp + ((size_t)(p0 + 0) * BATCH + b) * DIM + tid * VEC;
    async_issue_row(row0, &buf[0][tid * VEC]);

    const float scale = KD_T / (float)DIM / (float)PNUM;
    for (int i = 0; i < PCHUNK; ++i) {
        if (i + 1 < PCHUNK) {
            const float* rown = ebp + ((size_t)(p0 + i + 1) * BATCH + b) * DIM + tid * VEC;
            async_issue_row(rown, &buf[(i + 1) & 1][tid * VEC]);
            async_wait_keep2();            // row i landed; row i+1 in flight
        } else {
            async_wait_all();
        }

        const float4* e4 = (const float4*)&buf[i & 1][tid * VEC];
        const float4 e0 = e4[0], e1 = e4[1];
        const float e[VEC] = {e0.x, e0.y, e0.z, e0.w, e1.x, e1.y, e1.z, e1.w};

        float dg[VEC], db[VEC];
        #pragma unroll
        for (int k = 0; k < VEC; ++k) {
            const float xg = g[k] - e[k];
            const float xb = m[k] - e[k];
            dg[k] = xg * xg;
            db[k] = xb * xb;
        }

        const float r = sym_kl_row(dg, db, shm, shs);
        if (tid == 0) ws_dcl[(size_t)(p0 + i) * BATCH + b] = r * scale;
    }
}

// ---------------- kernel 4: final deterministic reduction ------------------
// dcl's last 64->1 stage runs through V_WMMA_F32_16X16X4_F32 with B = ones:
// D[m][n] = rowsum(A[m,:]); lane0 rows 0-7 + lane16 rows 8-15 = exact total.
__global__ void final_kernel(const float* __restrict__ ws_dil,
                             const float* __restrict__ ws_dcl,
                             float* __restrict__ out) {
    __shared__ float sh[TPB];
    __shared__ float red64[64];
    __shared__ float half_sums[32];
    const int tid = threadIdx.x;

    // soft_dil: 256 partials, fixed-order tree
    sh[tid] = ws_dil[tid];
    __syncthreads();
    for (int s = TPB / 2; s > 0; s >>= 1) {
        if (tid < s) sh[tid] += sh[tid + s];
        __syncthreads();
    }
    const float dil = sh[0];
    __syncthreads();

    // soft_dcl: 14336 partials -> 256 per-thread accs (fixed strided order)
    float acc = 0.f;
    for (int i = tid; i < PNUM * BATCH; i += TPB) acc += ws_dcl[i];
    sh[tid] = acc;
    __syncthreads();
    if (tid < 64)
        red64[tid] = sh[4 * tid] + sh[4 * tid + 1] + sh[4 * tid + 2] + sh[4 * tid + 3];
    __syncthreads();

    if (tid < 32) {   // wave 0 only; EXEC all 1s inside this wave-uniform branch
        v2f a;
        if (tid < 16) {             // A layout: lanes 0-15 hold K=0,1 for M=lane
            a.x = red64[4 * tid + 0];
            a.y = red64[4 * tid + 1];
        } else {                    // lanes 16-31 hold K=2,3 for M=lane-16
            a.x = red64[4 * (tid - 16) + 2];
            a.y = red64[4 * (tid - 16) + 3];
        }
        v2f ones; ones.x = 1.0f; ones.y = 1.0f;
        v8f c = {};
        v8f d = __builtin_amdgcn_wmma_f32_16x16x4_f32(
            /*neg_a=*/false, a, /*neg_b=*/false, ones,
            /*c_mod=*/(short)0, c, /*reuse_a=*/false, /*reuse_b=*/false);
        half_sums[tid] = d[0] + d[1] + d[2] + d[3] + d[4] + d[5] + d[6] + d[7];
    }
    __syncthreads();

    if (tid == 0) {
        out[0] = dil;
        out[1] = half_sums[0] + half_sums[16];  // rows 0-7 + rows 8-15
    }
}

// ---------------- launcher --------------------------------------------------
extern "C" void kernel_launch(void* const* d_in, const int* in_sizes, int n_in,
                              void* d_out, int out_size, void* d_ws, size_t ws_size,
                              hipStream_t stream) {
    const float* ebg = (const float*)d_in[0];
    const float* ebp = (const float*)d_in[1];
    // d_in[2] = labels (unused by the forward values)
    float* out = (float*)d_out;

    float* ws     = (float*)d_ws;
    float* bar    = ws;                          // BATCH*DIM floats (2 MB)
    float* ws_dil = bar + (size_t)BATCH * DIM;   // BATCH floats
    float* ws_dcl = ws_dil + BATCH;              // PNUM*BATCH floats

    bar_kernel  <<<BATCH * DIM / 4 / TPB, TPB, 0, stream>>>(ebp, bar);
    dil_kernel  <<<BATCH,                 TPB, 0, stream>>>(ebg, bar, ws_dil);
    dcl_kernel  <<<BATCH * SPLIT,         TPB, 0, stream>>>(ebg, bar, ebp, ws_dcl);
    final_kernel<<<1,                     TPB, 0, stream>>>(ws_dil, ws_dcl, out);
}